// _BaseFilter_66090956751486
// MI455X (gfx1250) — compile-verified
//
#include <hip/hip_runtime.h>
#include <cstdint>

// ---------------- problem geometry ----------------
#define N_ROW   65536
#define PADL    15
#define NEXT_   (N_ROW + 2*PADL)        // 65566 extended length
#define NT      256                     // threads per block (8 waves, wave32)
#define CHUNK   ((NEXT_ + NT - 1)/NT)   // 257 samples per thread
#define MSTR    17                      // padded stride for 4x4 matrices in LDS
#define VSTR    5                       // padded stride for state vectors in LDS

// ---------------- LDS layout (floats) ----------------
#define OFF_E    1                      // e[PADL] lands 64B-aligned
#define OFF_MSA  65568
#define OFF_MSB  (OFF_MSA + NT*MSTR)
#define OFF_VSA  (OFF_MSB + NT*MSTR)
#define OFF_VSB  (OFF_VSA + NT*VSTR)
#define OFF_RED  (OFF_VSB + NT*VSTR)
#define SMEM_FLOATS (OFF_RED + NT)      // 77088 floats = 308352 B  (< 320 KB WGP LDS)

typedef __attribute__((ext_vector_type(2))) float v2f;
typedef __attribute__((ext_vector_type(8))) float v8f;
typedef __attribute__((ext_vector_type(4))) unsigned int v4u;
typedef __attribute__((ext_vector_type(4))) int v4i;
typedef __attribute__((ext_vector_type(8))) int v8i;

#if defined(__has_builtin)
#  if __has_builtin(__builtin_amdgcn_tensor_load_to_lds) && \
      __has_builtin(__builtin_amdgcn_tensor_store_from_lds) && \
      __has_builtin(__builtin_amdgcn_s_wait_tensorcnt)
#    define HAVE_TDM 1
#  endif
#  if __has_builtin(__builtin_amdgcn_wmma_f32_16x16x4_f32)
#    define HAVE_WMMA_F32X4 1
#  endif
#endif

// 4x4 row-major matrix multiply: O = X * Y
__device__ inline void mmul4(const float* X, const float* Y, float* O) {
#pragma unroll
  for (int r = 0; r < 4; ++r) {
#pragma unroll
    for (int c = 0; c < 4; ++c) {
      float s = X[r*4+0] * Y[0*4+c];
      s = fmaf(X[r*4+1], Y[1*4+c], s);
      s = fmaf(X[r*4+2], Y[2*4+c], s);
      s = fmaf(X[r*4+3], Y[3*4+c], s);
      O[r*4+c] = s;
    }
  }
}

#ifdef HAVE_TDM
// Build TDM D# groups 0/1 for a contiguous 65536-float tile viewed as 4096x16,
// moved between `lds_byte` and `ga` (per CDNA5 ISA).
__device__ inline void tdm_desc(unsigned lds_byte, unsigned long long ga,
                                v4u& g0, v8i& g1) {
  g0[0] = 1u;                                   // count=1, is_restore=0, no gather
  g0[1] = lds_byte;                             // lds_addr
  g0[2] = (unsigned)(ga & 0xFFFFFFFFull);       // global_addr[31:0]
  g0[3] = ((unsigned)((ga >> 32) & 0x01FFFFFFull)) | (2u << 30); // ga[56:32] | type=2
  g1[0] = (2 << 16);                            // workgroup_mask=0, data_size=4B
  g1[1] = (int)((4096u & 0xFFFFu) << 16);       // tensor_dim0[15:0] (=4096)
  g1[2] = (int)((4096u >> 16) | ((16u & 0xFFFFu) << 16)); // dim0 hi | tensor_dim1 lo (=16)
  g1[3] = (int)((16u >> 16) | (4096u << 16));   // dim1 hi | tile_dim0 (=4096)
  g1[4] = 16;                                   // tile_dim1=16, tile_dim2=0
  g1[5] = 4096;                                 // tensor_dim0_stride[31:0]
  g1[6] = 0;                                    // stride hi bits
  g1[7] = 0;
}
#endif

// One lfilter pass over the extended signal in LDS (in place), using a
// 256-way chunked affine scan of the DF2T 4-state recurrence.
// DIR=+1: forward in time.  DIR=-1: reversed (time-flipped) pass.
template<int DIR>
__device__ void iir_pass(float* __restrict__ smem, int tid,
                         float b0, float b1, float b2, float b3, float b4,
                         float na1, float na2, float na3, float na4,
                         float inscale, float outscale) {
  float* e   = smem + OFF_E;
  float* MsA = smem + OFF_MSA;
  float* MsB = smem + OFF_MSB;
  float* VsA = smem + OFF_VSA;
  float* VsB = smem + OFF_VSB;

  const int s0  = tid * CHUNK;
  const int len = (s0 + CHUNK <= NEXT_) ? CHUNK : (NEXT_ - s0);

  // ---- Phase A: zero-state chunk response v = final state with z_init = 0
  float z0 = 0.f, z1 = 0.f, z2 = 0.f, z3 = 0.f;
  for (int k = 0; k < len; ++k) {
    const int pos = (DIR > 0) ? (s0 + k) : (NEXT_ - 1 - (s0 + k));
    const float xv = e[pos] * inscale;
    const float y  = fmaf(b0, xv, z0);
    z0 = fmaf(na1, y, fmaf(b1, xv, z1));
    z1 = fmaf(na2, y, fmaf(b2, xv, z2));
    z2 = fmaf(na3, y, fmaf(b3, xv, z3));
    z3 = fmaf(na4, y, b4 * xv);
  }

  // ---- Phase B: homogeneous transition matrix M = A^len (binary powering)
  float Am[16] = { na1, 1.f, 0.f, 0.f,
                   na2, 0.f, 1.f, 0.f,
                   na3, 0.f, 0.f, 1.f,
                   na4, 0.f, 0.f, 0.f };
  float P[16]  = { 1.f,0.f,0.f,0.f, 0.f,1.f,0.f,0.f, 0.f,0.f,1.f,0.f, 0.f,0.f,0.f,1.f };
  int ee = len;
  while (ee) {
    if (ee & 1) { float t[16]; mmul4(P, Am, t);
#pragma unroll
      for (int i = 0; i < 16; ++i) P[i] = t[i]; }
    ee >>= 1;
    if (ee) { float t[16]; mmul4(Am, Am, t);
#pragma unroll
      for (int i = 0; i < 16; ++i) Am[i] = t[i]; }
  }
#pragma unroll
  for (int i = 0; i < 16; ++i) MsA[tid*MSTR + i] = P[i];
  VsA[tid*VSTR + 0] = z0; VsA[tid*VSTR + 1] = z1;
  VsA[tid*VSTR + 2] = z2; VsA[tid*VSTR + 3] = z3;
  __syncthreads();

  // ---- Phase C: Hillis-Steele inclusive scan of affines (M,v):
  // (M_t, v_t) ∘ (M_p, v_p) = (M_t*M_p, M_t*v_p + v_t), identity when t < step.
  float* Min = MsA; float* Mout = MsB;
  float* Vin = VsA; float* Vout = VsB;
  for (int s = 1; s < NT; s <<= 1) {
#ifdef HAVE_WMMA_F32X4
    // Batched 4x4 composition via V_WMMA_F32_16X16X4_F32: per wave, 8 WMMAs
    // each compute 4 products as the diagonal blocks of (16x4)x(4x16).
    {
      const int lane  = tid & 31;
      const int wb    = tid & ~31;
      const int mrow  = lane & 15;          // A-row / B-col index (0..15)
      const int khalf = (lane >> 4) << 1;   // K pair: lanes 0-15 -> {0,1}, 16-31 -> {2,3}
      for (int g = 0; g < 8; ++g) {
        const int q0 = wb + 4*g;            // 4 threads whose matrices this WMMA composes
        const int ta = q0 + (mrow >> 2);    // owner of stacked A row / B column group
        const int ra = mrow & 3;            // row within 4x4 (A side)
        const int cb = mrow & 3;            // col within 4x4 (B side)
        v2f af; af.x = Min[ta*MSTR + ra*4 + khalf];
                af.y = Min[ta*MSTR + ra*4 + khalf + 1];
        const int pp = ta - s;              // scan partner (identity if negative)
        v2f bf;
        bf.x = (pp >= 0) ? Min[pp*MSTR + khalf*4 + cb]     : ((khalf     == cb) ? 1.f : 0.f);
        bf.y = (pp >= 0) ? Min[pp*MSTR + (khalf+1)*4 + cb] : ((khalf + 1 == cb) ? 1.f : 0.f);
        v8f cz = {0.f,0.f,0.f,0.f,0.f,0.f,0.f,0.f};
        v8f d = __builtin_amdgcn_wmma_f32_16x16x4_f32(
            false, af, false, bf, (short)0, cz, false, false);
        // Diagonal-block extraction. Lanes 0-3 & 24-27 need d[0..3] (blocks
        // j=0,2); lanes 4-7 & 28-31 need d[4..7] (blocks j=1,3). Constant
        // indices + one predicate avoid dynamic-extract cndmask chains.
        const bool active = (lane < 8) | (lane >= 24);
        if (active) {
          const int j   = (lane < 8) ? (lane >> 2) : ((lane - 16) >> 2);
          const int col = lane & 3;
          const int tq  = q0 + j;
          const bool hi = (lane & 4) != 0;
          const float r0 = hi ? d[4] : d[0];
          const float r1 = hi ? d[5] : d[1];
          const float r2 = hi ? d[6] : d[2];
          const float r3 = hi ? d[7] : d[3];
          Mout[tq*MSTR + 0*4 + col] = r0;
          Mout[tq*MSTR + 1*4 + col] = r1;
          Mout[tq*MSTR + 2*4 + col] = r2;
          Mout[tq*MSTR + 3*4 + col] = r3;
        }
      }
    }
#else
    {
      float Mo[16], Mp[16], Mn[16];
#pragma unroll
      for (int i = 0; i < 16; ++i) Mo[i] = Min[tid*MSTR + i];
      const int p = tid - s;
      if (p >= 0) {
#pragma unroll
        for (int i = 0; i < 16; ++i) Mp[i] = Min[p*MSTR + i];
      } else {
#pragma unroll
        for (int i = 0; i < 16; ++i) Mp[i] = (i % 5 == 0) ? 1.f : 0.f;
      }
      mmul4(Mo, Mp, Mn);
#pragma unroll
      for (int i = 0; i < 16; ++i) Mout[tid*MSTR + i] = Mn[i];
    }
#endif
    // v composition (scalar, 16 FMAs per thread)
    {
      float Mo[16];
#pragma unroll
      for (int i = 0; i < 16; ++i) Mo[i] = Min[tid*MSTR + i];
      const float vo0 = Vin[tid*VSTR+0], vo1 = Vin[tid*VSTR+1];
      const float vo2 = Vin[tid*VSTR+2], vo3 = Vin[tid*VSTR+3];
      const int p = tid - s;
      float vp0 = 0.f, vp1 = 0.f, vp2 = 0.f, vp3 = 0.f;
      if (p >= 0) { vp0 = Vin[p*VSTR+0]; vp1 = Vin[p*VSTR+1];
                    vp2 = Vin[p*VSTR+2]; vp3 = Vin[p*VSTR+3]; }
      Vout[tid*VSTR+0] = fmaf(Mo[3],  vp3, fmaf(Mo[2],  vp2, fmaf(Mo[1],  vp1, fmaf(Mo[0],  vp0, vo0))));
      Vout[tid*VSTR+1] = fmaf(Mo[7],  vp3, fmaf(Mo[6],  vp2, fmaf(Mo[5],  vp1, fmaf(Mo[4],  vp0, vo1))));
      Vout[tid*VSTR+2] = fmaf(Mo[11], vp3, fmaf(Mo[10], vp2, fmaf(Mo[9],  vp1, fmaf(Mo[8],  vp0, vo2))));
      Vout[tid*VSTR+3] = fmaf(Mo[15], vp3, fmaf(Mo[14], vp2, fmaf(Mo[13], vp1, fmaf(Mo[12], vp0, vo3))));
    }
    __syncthreads();
    float* t;
    t = Min; Min = Mout; Mout = t;
    t = Vin; Vin = Vout; Vout = t;
  }
  // After 8 rounds results sit in the A buffers (Vin == VsA).
  // z_init(t) = scanned v of thread t-1 applied to zero global initial state.
  z0 = z1 = z2 = z3 = 0.f;
  if (tid > 0) {
    z0 = Vin[(tid-1)*VSTR+0]; z1 = Vin[(tid-1)*VSTR+1];
    z2 = Vin[(tid-1)*VSTR+2]; z3 = Vin[(tid-1)*VSTR+3];
  }

  // ---- Phase D: final chunk pass with correct initial state, write in place
  for (int k = 0; k < len; ++k) {
    const int pos = (DIR > 0) ? (s0 + k) : (NEXT_ - 1 - (s0 + k));
    const float xv = e[pos] * inscale;
    const float y  = fmaf(b0, xv, z0);
    z0 = fmaf(na1, y, fmaf(b1, xv, z1));
    z1 = fmaf(na2, y, fmaf(b2, xv, z2));
    z2 = fmaf(na3, y, fmaf(b3, xv, z3));
    z3 = fmaf(na4, y, b4 * xv);
    e[pos] = y * outscale;
  }
  __syncthreads();
}

__global__ void filtfilt_rows_kernel(const float* __restrict__ x,
                                     const float* __restrict__ bc,
                                     const float* __restrict__ ac,
                                     float* __restrict__ out) {
  extern __shared__ float smem[];
  float* e   = smem + OFF_E;
  float* red = smem + OFF_RED;

  const int tid = (int)threadIdx.x;
  const int row = (int)blockIdx.x;
  const float* xrow = x   + (size_t)row * N_ROW;
  float*       orow = out + (size_t)row * N_ROW;

  const float b0 = bc[0], b1 = bc[1], b2 = bc[2], b3 = bc[3], b4 = bc[4];
  const float na1 = -ac[1], na2 = -ac[2], na3 = -ac[3], na4 = -ac[4];

  // ---- stage row into LDS e[PADL .. PADL+N) via Tensor Data Mover ----
#ifdef HAVE_TDM
  if (tid < 32) {   // one wave issues the TDM op (EXEC ignored by TDM anyway)
    const unsigned ldsb = (unsigned)(uintptr_t)(void*)&e[PADL];
    v4u g0; v8i g1;
    tdm_desc(ldsb, (unsigned long long)(uintptr_t)xrow, g0, g1);
    v4i gz; gz[0] = 0; gz[1] = 0; gz[2] = 0; gz[3] = 0;
    v8i gz8;
#pragma unroll
    for (int i = 0; i < 8; ++i) gz8[i] = 0;
    __builtin_amdgcn_tensor_load_to_lds(g0, g1, gz, gz, gz8, 0);
    __builtin_amdgcn_s_wait_tensorcnt(0);
  }
#else
  for (int i = tid; i < N_ROW/4; i += NT)
    ((float4*)(e + PADL))[i] = ((const float4*)xrow)[i];
#endif
  __syncthreads();

  // ---- per-row scale = max|x| (block reduction) + odd extension boundaries ----
  float m = 0.f;
  for (int i = tid; i < N_ROW; i += NT) m = fmaxf(m, fabsf(e[PADL + i]));
  if (tid < PADL) {
    const float x0 = e[PADL], xl = e[PADL + N_ROW - 1];
    e[PADL - 1 - tid]     = 2.f * x0 - e[PADL + 1 + tid];
    e[PADL + N_ROW + tid] = 2.f * xl - e[PADL + N_ROW - 2 - tid];
  }
  red[tid] = m;
  __syncthreads();
  for (int s = NT/2; s > 0; s >>= 1) {
    if (tid < s) red[tid] = fmaxf(red[tid], red[tid + s]);
    __syncthreads();
  }
  const float scale = red[0];
  const float inv   = 1.f / scale;
  __syncthreads();

  // ---- filtfilt: forward pass (normalized), backward pass (rescaled) ----
  iir_pass<+1>(smem, tid, b0,b1,b2,b3,b4, na1,na2,na3,na4, inv, 1.f);
  iir_pass<-1>(smem, tid, b0,b1,b2,b3,b4, na1,na2,na3,na4, 1.f, scale);

  // ---- store cropped region e[PADL .. PADL+N) to out via TDM ----
#ifdef HAVE_TDM
  if (tid < 32) {
    const unsigned ldsb = (unsigned)(uintptr_t)(void*)&e[PADL];
    v4u g0; v8i g1;
    tdm_desc(ldsb, (unsigned long long)(uintptr_t)orow, g0, g1);
    v4i gz; gz[0] = 0; gz[1] = 0; gz[2] = 0; gz[3] = 0;
    v8i gz8;
#pragma unroll
    for (int i = 0; i < 8; ++i) gz8[i] = 0;
    __builtin_amdgcn_tensor_store_from_lds(g0, g1, gz, gz, gz8, 0);
    __builtin_amdgcn_s_wait_tensorcnt(0);
  }
#else
  for (int i = tid; i < N_ROW/4; i += NT)
    ((float4*)orow)[i] = ((const float4*)(e + PADL))[i];
#endif
}

extern "C" void kernel_launch(void* const* d_in, const int* in_sizes, int n_in,
                              void* d_out, int out_size, void* d_ws, size_t ws_size,
                              hipStream_t stream) {
  (void)n_in; (void)out_size; (void)d_ws; (void)ws_size;
  const float* x = (const float*)d_in[0];
  const float* b = (const float*)d_in[1];
  const float* a = (const float*)d_in[2];
  float* out = (float*)d_out;
  const int rows = in_sizes[0] / N_ROW;            // 8*32 = 256 rows
  const size_t shmem = (size_t)SMEM_FLOATS * sizeof(float);  // ~301 KB of WGP LDS
  filtfilt_rows_kernel<<<dim3(rows), dim3(NT), shmem, stream>>>(x, b, a, out);
}